// CascadeGDCN0_17162689315366
// MI455X (gfx1250) — compile-verified
//
#include <hip/hip_runtime.h>
#include <hip/hip_bf16.h>

// CascadeGDCN for MI455X (gfx1250, wave32, WMMA).
// Pipeline per call:
//   accum = 0
//   for k in 0..K-1:
//     X = (max(d_out,EPS) ⊙ H) @ theta_out[k]        [WMMA f32 16x16x4]
//     accum[rows] += alpha_k * vals * X[cols]          [global_atomic_add_f32]
//     X = (max(d_in ,EPS) ⊙ H) @ theta_in[k]
//     accum[cols] += alpha_k * vals * X[rows]
//   out = sigmoid(accum @ Theta) + H                   [WMMA f32 + epilogue]
// Workspace: X (N*64 f32) + accum (N*64 f32) = 51.2 MB — both L2-resident.

typedef __attribute__((ext_vector_type(2))) float v2f;
typedef __attribute__((ext_vector_type(8))) float v8f;

#define EPSV 1e-8f

// MODE 0: A = max(dvec,EPS) ⊙ Hmat; store raw result.
// MODE 1: A = Hmat (accum); epilogue out = sigmoid(c) + Hadd.
template <int MODE>
__global__ void __launch_bounds__(256)
gemm64_wmma(const float* __restrict__ Amat, const float* __restrict__ dvec,
            const float* __restrict__ W,    const float* __restrict__ Hadd,
            float* __restrict__ out, int N)
{
    __shared__ float sW[64 * 64];           // 16 KB of the 320 KB/WGP LDS
    const int t = threadIdx.x;

    // Cooperative load of the 64x64 weight matrix into LDS (float4).
    {
        const float4* W4  = (const float4*)W;
        float4*       sW4 = (float4*)sW;
#pragma unroll
        for (int i = 0; i < 4; ++i)
            sW4[t + i * 256] = W4[t + i * 256];
    }
    __syncthreads();

    const int wave  = t >> 5;               // 8 waves / block (wave32)
    const int lane  = t & 31;
    const int l     = lane & 15;            // column / M-row selector
    const int khalf = (lane >> 4) << 1;     // 0 or 2 : K-half per A/B layout
    const long tile_m = (long)blockIdx.x * 8 + wave;
    const long m0     = tile_m * 16;
    if (m0 >= N) return;                    // wave-uniform: EXEC stays all-ones

    const long rowA = (m0 + l < N) ? (m0 + l) : (N - 1);   // clamp (N%16==0 here)

    float dscale = 1.0f;
    if (MODE == 0) {
        float d = dvec[rowA];
        dscale = (d > EPSV) ? d : EPSV;     // max(degree, EPS)
    }

    v8f acc[4] = { {}, {}, {}, {} };        // 16x64 output strip per wave
    const float* Ap = Amat + rowA * 64;

#pragma unroll
    for (int kk = 0; kk < 64; kk += 4) {
        // A fragment (16x4 f32): VGPR0 holds K=khalf, VGPR1 holds K=khalf+1
        v2f a;
        a.x = dscale * Ap[kk + khalf + 0];
        a.y = dscale * Ap[kk + khalf + 1];
#pragma unroll
        for (int n = 0; n < 4; ++n) {
            // B fragment (4x16 f32): row K striped across lanes within a VGPR
            v2f b;
            b.x = sW[(kk + khalf + 0) * 64 + n * 16 + l];
            b.y = sW[(kk + khalf + 1) * 64 + n * 16 + l];
            acc[n] = __builtin_amdgcn_wmma_f32_16x16x4_f32(
                false, a, false, b, (short)0, acc[n], false, false);
        }
    }

    // D layout: VGPR r -> M = r (lanes 0-15) / r+8 (lanes 16-31), N = l
    const int mhalf = (lane >> 4) << 3;     // 0 or 8
#pragma unroll
    for (int r = 0; r < 8; ++r) {
        const long row = m0 + r + mhalf;
        if (row < N) {
#pragma unroll
            for (int n = 0; n < 4; ++n) {
                const long idx = row * 64 + n * 16 + l;
                float c = acc[n][r];
                if (MODE == 1)
                    out[idx] = 1.0f / (1.0f + __expf(-c)) + Hadd[idx];
                else
                    out[idx] = c;
            }
        }
    }
}

// accum[dst[e]] += alpha_k * vals[e] * X[src[e]]  — 16 lanes per edge, float4.
__global__ void __launch_bounds__(256)
scatter_spmm(const int* __restrict__ dstIdx, const int* __restrict__ srcIdx,
             const float* __restrict__ vals, const float* __restrict__ X,
             float* __restrict__ accum, const float* __restrict__ hop,
             int K, int k, int E)
{
    const long gid = (long)blockIdx.x * blockDim.x + threadIdx.x;
    const long e   = gid >> 4;
    if (e >= E) return;
    const int c = (int)(gid & 15) * 4;

    // alpha_k = softmax(hop_attention)[k]  (K is tiny; scalar loads)
    float mx = hop[0];
    for (int j = 1; j < K; ++j) mx = fmaxf(mx, hop[j]);
    float s = 0.0f;
    for (int j = 0; j < K; ++j) s += __expf(hop[j] - mx);
    const float alpha = __expf(hop[k] - mx) / s;

    const int  d = dstIdx[e];
    const int  sr = srcIdx[e];
    const float v = alpha * vals[e];

    const float4 x = *(const float4*)(X + (long)sr * 64 + c);
    float* p = accum + (long)d * 64 + c;
    unsafeAtomicAdd(p + 0, v * x.x);   // global_atomic_add_f32, L2-resident
    unsafeAtomicAdd(p + 1, v * x.y);
    unsafeAtomicAdd(p + 2, v * x.z);
    unsafeAtomicAdd(p + 3, v * x.w);
}

extern "C" void kernel_launch(void* const* d_in, const int* in_sizes, int n_in,
                              void* d_out, int out_size, void* d_ws, size_t ws_size,
                              hipStream_t stream)
{
    const float* H          = (const float*)d_in[0];   // [N,64]
    const float* edge_vals  = (const float*)d_in[1];   // [K,E]
    const float* out_degree = (const float*)d_in[2];   // [N]
    const float* in_degree  = (const float*)d_in[3];   // [N]
    const float* hop        = (const float*)d_in[4];   // [K]
    const float* Theta      = (const float*)d_in[5];   // [64,64]
    const float* theta_out  = (const float*)d_in[6];   // [K,64,64]
    const float* theta_in   = (const float*)d_in[7];   // [K,64,64]
    const int*   edge_index = (const int*)d_in[8];     // [K,2,E]

    const int  N = in_sizes[2];
    const int  K = in_sizes[4];
    const long E = (long)in_sizes[1] / K;

    float* X     = (float*)d_ws;                 // N*64 f32
    float* accum = X + (size_t)N * 64;           // N*64 f32
    float* outp  = (float*)d_out;

    hipMemsetAsync(accum, 0, (size_t)N * 64 * sizeof(float), stream);

    const int tilesM = (N + 15) / 16;
    dim3 gemmGrid((tilesM + 7) / 8);
    dim3 blk(256);
    dim3 scGrid((unsigned)((E * 16 + 255) / 256));

    for (int k = 0; k < K; ++k) {
        const int* rows = edge_index + (size_t)k * 2 * E;
        const int* cols = rows + E;
        const float* vk = edge_vals + (size_t)k * E;

        // out-direction: accum[rows] += a_k * v * ((d_out⊙H)@theta_out[k])[cols]
        gemm64_wmma<0><<<gemmGrid, blk, 0, stream>>>(
            H, out_degree, theta_out + (size_t)k * 64 * 64, nullptr, X, N);
        scatter_spmm<<<scGrid, blk, 0, stream>>>(rows, cols, vk, X, accum,
                                                 hop, K, k, (int)E);

        // in-direction (A^T): accum[cols] += a_k * v * ((d_in⊙H)@theta_in[k])[rows]
        gemm64_wmma<0><<<gemmGrid, blk, 0, stream>>>(
            H, in_degree, theta_in + (size_t)k * 64 * 64, nullptr, X, N);
        scatter_spmm<<<scGrid, blk, 0, stream>>>(cols, rows, vk, X, accum,
                                                 hop, K, k, (int)E);
    }

    // out = sigmoid(accum @ Theta) + H
    gemm64_wmma<1><<<gemmGrid, blk, 0, stream>>>(
        accum, nullptr, Theta, H, outp, N);
}